// GraphConvLayer_29420525978023
// MI455X (gfx1250) — compile-verified
//
#include <hip/hip_runtime.h>
#include <hip/hip_bf16.h>

typedef __attribute__((ext_vector_type(2))) float v2f;
typedef __attribute__((ext_vector_type(8))) float v8f;

#define F_DIM 128
#define B_DIM 2

// ---------------- zero init ----------------
__global__ void zero_f4_kernel(float4* __restrict__ p, int n4) {
    int i = blockIdx.x * blockDim.x + threadIdx.x;
    if (i < n4) p[i] = make_float4(0.f, 0.f, 0.f, 0.f);
}

__global__ void zero_f_kernel(float* __restrict__ p, int n) {
    int i = blockIdx.x * blockDim.x + threadIdx.x;
    if (i < n) p[i] = 0.f;
}

// ---------------- degree: deg[dst] += 1 ----------------
__global__ void degree_kernel(const int* __restrict__ ei, float* __restrict__ deg, int E) {
    int e = blockIdx.x * blockDim.x + threadIdx.x;
    if (e < E) atomicAdd(&deg[ei[E + e]], 1.0f);
}

// ---------------- deg -> clip(deg,1)^-0.5 in place ----------------
__global__ void dinv_kernel(float* __restrict__ deg, int N) {
    int i = blockIdx.x * blockDim.x + threadIdx.x;
    if (i < N) {
        float d = deg[i];
        d = fmaxf(d, 1.0f);
        deg[i] = 1.0f / sqrtf(d);
    }
}

// ---------------- scatter: agg[b,dst,:] += x[b,src,:] * norm ----------------
// one wave32 per edge; lane owns a contiguous float4 chunk (coalesced 512B/row)
__global__ void scatter_kernel(const float* __restrict__ x,
                               const int* __restrict__ ei,
                               const float* __restrict__ ew,
                               const float* __restrict__ dinv,
                               float* __restrict__ agg,
                               int E, int N) {
    const int wid  = (blockIdx.x * blockDim.x + threadIdx.x) >> 5;
    const int lane = threadIdx.x & 31;
    if (wid >= E) return;

    const int   s    = ei[wid];
    const int   d    = ei[E + wid];
    const float norm = dinv[s] * ew[wid] * dinv[d];

#pragma unroll
    for (int b = 0; b < B_DIM; ++b) {
        const float4 v = ((const float4*)(x + ((size_t)b * N + s) * F_DIM))[lane];
        float* dp = agg + ((size_t)b * N + d) * F_DIM + lane * 4;
        atomicAdd(dp + 0, v.x * norm);
        atomicAdd(dp + 1, v.y * norm);
        atomicAdd(dp + 2, v.z * norm);
        atomicAdd(dp + 3, v.w * norm);
    }
}

// ---------------- GEMM: out[M,128] = agg[M,128] @ W[128,128]^T + bias ----------------
// One wave32 per 16x16 output tile, V_WMMA_F32_16X16X4_F32, K looped 128/4 = 32 times.
// f32 WMMA VGPR layouts (ISA 7.12.2):
//   A 16x4 : lane<16 -> M=lane,   K={k,k+1};  lane>=16 -> M=lane-16, K={k+2,k+3}
//   B 4x16 : lane<16 -> N=lane,   K={k,k+1};  lane>=16 -> N=lane-16, K={k+2,k+3}
//            (B[k][n] = W[n][k], so indexing form matches A with row->out col)
//   C/D    : VGPR r -> row (r + 8*half), col = lane&15
__global__ void gemm_wmma_kernel(const float* __restrict__ agg,
                                 const float* __restrict__ W,
                                 const float* __restrict__ bias,
                                 float* __restrict__ out,
                                 int M) {
    const int wave = (blockIdx.x * blockDim.x + threadIdx.x) >> 5;
    const int lane = threadIdx.x & 31;
    const int tiles_n = F_DIM / 16;           // 8
    const int tile_m = wave / tiles_n;
    const int tile_n = wave % tiles_n;
    if (tile_m * 16 >= M) return;             // whole-wave uniform: EXEC stays all-1s

    const int row_base = tile_m * 16;
    const int col_base = tile_n * 16;
    const int half = lane >> 4;               // 0 or 1
    const int lid  = lane & 15;               // 0..15

    const float* arow = agg + (size_t)(row_base + lid) * F_DIM;
    const float* wrow = W   + (size_t)(col_base + lid) * F_DIM;

    v8f c = {};
#pragma unroll
    for (int k = 0; k < F_DIM; k += 4) {
        const int ka = k + half * 2;
        const float2 av = *(const float2*)(arow + ka);
        const float2 bv = *(const float2*)(wrow + ka);
        v2f a; a.x = av.x; a.y = av.y;
        v2f b; b.x = bv.x; b.y = bv.y;
        c = __builtin_amdgcn_wmma_f32_16x16x4_f32(
                /*neg_a=*/false, a, /*neg_b=*/false, b,
                /*c_mod=*/(short)0, c, /*reuse_a=*/false, /*reuse_b=*/false);
    }

    const float bb = bias[col_base + lid];
#pragma unroll
    for (int r = 0; r < 8; ++r) {
        const int m = row_base + r + half * 8;
        out[(size_t)m * F_DIM + col_base + lid] = c[r] + bb;
    }
}

extern "C" void kernel_launch(void* const* d_in, const int* in_sizes, int n_in,
                              void* d_out, int out_size, void* d_ws, size_t ws_size,
                              hipStream_t stream) {
    const float* x    = (const float*)d_in[0];   // [B,N,128]
    const int*   ei   = (const int*)d_in[1];     // [2,E]
    const float* ew   = (const float*)d_in[2];   // [E]
    const float* W    = (const float*)d_in[3];   // [128,128]
    const float* bias = (const float*)d_in[4];   // [128]
    float*       out  = (float*)d_out;           // [B,N,128]

    const int E = in_sizes[2];                       // edge count
    const int N = in_sizes[0] / (B_DIM * F_DIM);     // nodes
    const int M = B_DIM * N;                         // GEMM rows

    // workspace layout: agg [M*128] floats, then deg [N] floats
    float* agg = (float*)d_ws;
    float* deg = agg + (size_t)M * F_DIM;

    // 1) zero agg (vectorized) + deg
    {
        int n4 = (M * F_DIM) / 4;
        zero_f4_kernel<<<(n4 + 255) / 256, 256, 0, stream>>>((float4*)agg, n4);
        zero_f_kernel<<<(N + 255) / 256, 256, 0, stream>>>(deg, N);
    }
    // 2) degrees
    degree_kernel<<<(E + 255) / 256, 256, 0, stream>>>(ei, deg, E);
    // 3) deg -> clip(deg,1)^-0.5
    dinv_kernel<<<(N + 255) / 256, 256, 0, stream>>>(deg, N);
    // 4) normalized scatter-add (wave per edge)
    {
        int waves = E;
        int blocks = (waves + 7) / 8;                // 8 waves (256 threads) per block
        scatter_kernel<<<blocks, 256, 0, stream>>>(x, ei, ew, deg, agg, E, N);
    }
    // 5) WMMA GEMM + bias
    {
        int tiles_m = (M + 15) / 16;
        int waves = tiles_m * (F_DIM / 16);
        int blocks = (waves + 7) / 8;
        gemm_wmma_kernel<<<blocks, 256, 0, stream>>>(agg, W, bias, out, M);
    }
}